// Model_70471823392991
// MI455X (gfx1250) — compile-verified
//
#include <hip/hip_runtime.h>
#include <hip/hip_bf16.h>
#include <stdint.h>

// MoE expert-count histogram for MI455X (gfx1250, wave32).
// 134 MB streamed @ 23.3 TB/s => ~5.8 us floor; pure bandwidth problem.
// CDNA5 path: 4-deep global_load_async_to_lds_b128 pipeline + s_wait_asynccnt.
// Hot loop does exactly ONE ds_add_u32 per id (256-bin global-id histogram
// per wave); expert_map is applied once per bin during the reduction.

#define TPB        256
#define WAVES      (TPB / 32)          // wave32 on gfx1250
#define PIPE       4                   // async pipeline depth (4 x 4KB staging)
#define MAX_BINS   256                 // global-id bins
#define MAX_MAP    256

__global__ void Model_zero_out(int* __restrict__ out, int n) {
    int i = blockIdx.x * blockDim.x + threadIdx.x;
    if (i < n) out[i] = 0;
}

__global__ __launch_bounds__(TPB)
void Model_moe_hist(const int* __restrict__ ids,
                    unsigned int nvec,          // number of int4 (16B) vectors
                    int ntail,                  // leftover scalar ids
                    const int* __restrict__ expert_map,
                    int map_size,
                    int nbins,
                    int* __restrict__ out) {
    __shared__ int4 stage[PIPE][TPB];           // 16 KB async staging
    __shared__ int  smap[MAX_MAP];              //  1 KB global->local map
    __shared__ int  whist[WAVES][MAX_BINS];     //  8 KB per-wave global-id hist

    const int tid = threadIdx.x;
    const int wid = tid >> 5;                   // wave32

    if (tid < MAX_MAP)
        smap[tid] = (tid < map_size) ? expert_map[tid] : -1;
    for (int i = tid; i < WAVES * MAX_BINS; i += TPB)
        ((int*)whist)[i] = 0;
    __syncthreads();

    const unsigned stride = gridDim.x * TPB;
    const unsigned idx0   = blockIdx.x * TPB + tid;
    const unsigned iters  = (nvec + stride - 1u) / stride;   // uniform across grid
    const uint64_t base   = (uint64_t)(uintptr_t)ids;

    // Issue one async 16B global->LDS copy for tile k into buffer k%PIPE.
    // Addresses are clamped, never predicated: EXEC stays all-1s so every
    // wave's ASYNCcnt bookkeeping is uniform and in-order.
    auto issue = [&](unsigned k) {
        unsigned idx = idx0 + k * stride;
        if (idx >= nvec) idx = nvec - 1u;       // clamp; guarded at use
        unsigned voff = idx * 16u;              // byte offset (fits 31 bits)
        unsigned lds  = (unsigned)(uintptr_t)&stage[k % PIPE][tid];
        asm volatile("global_load_async_to_lds_b128 %0, %1, %2"
                     :: "v"(lds), "v"(voff), "s"(base)
                     : "memory");
    };

    if (iters > 0u) {
#pragma unroll
        for (unsigned p = 0; p < PIPE; ++p)
            if (p < iters) issue(p);

        for (unsigned k = 0; k < iters; ++k) {
            // In-order async completion: with PIPE outstanding, cnt<=PIPE-1
            // guarantees tile k has landed in LDS. Epilogue drains fully.
            if (k + PIPE <= iters)
                asm volatile("s_wait_asynccnt 3" ::: "memory");
            else
                asm volatile("s_wait_asynccnt 0" ::: "memory");

            int4 v = stage[k % PIPE][tid];      // ds_load_b128 of own slot
            // Drain the LDS read before the async engine may overwrite
            // this slot with tile k+PIPE.
            asm volatile("s_wait_dscnt 0" ::: "memory");

            if (k + PIPE < iters) issue(k + PIPE);

            unsigned idx = idx0 + k * stride;
            if (idx < nvec) {
                int e[4] = {v.x, v.y, v.z, v.w};
#pragma unroll
                for (int j = 0; j < 4; ++j) {
                    unsigned g = (unsigned)e[j];         // negative -> huge
                    if (g < (unsigned)MAX_BINS)
                        atomicAdd(&whist[wid][g], 1);    // ds_add_u32 (1/id)
                }
            }
        }
    }
    __syncthreads();

    // Reduce per-wave global-id hist, apply expert_map once per bin,
    // accumulate into d_out with global atomics.
    for (int b = tid; b < MAX_BINS; b += TPB) {
        int s = 0;
#pragma unroll
        for (int w = 0; w < WAVES; ++w) s += whist[w][b];
        int lid = smap[b];
        if (s && (unsigned)lid < (unsigned)nbins)
            atomicAdd(&out[lid], s);            // global_atomic_add_u32
    }

    // Scalar tail (N % 4 ids), one thread of block 0.
    if (blockIdx.x == 0 && tid == 0) {
        for (int t = 0; t < ntail; ++t) {
            unsigned g = (unsigned)ids[(size_t)nvec * 4u + t];
            int lid = (g < (unsigned)MAX_MAP) ? smap[g] : -1;
            if ((unsigned)lid < (unsigned)nbins)
                atomicAdd(&out[lid], 1);
        }
    }
}

extern "C" void kernel_launch(void* const* d_in, const int* in_sizes, int n_in,
                              void* d_out, int out_size, void* d_ws, size_t ws_size,
                              hipStream_t stream) {
    const int* ids        = (const int*)d_in[0];   // topk_ids, int32
    // d_in[1] = num_local_experts (device scalar) -- equals out_size, unused
    const int* expert_map = (const int*)d_in[2];   // int32[256]

    const long long n   = (long long)in_sizes[0];
    const int map_size  = (n_in > 2) ? in_sizes[2] : 0;
    const unsigned nvec = (unsigned)(n / 4);
    const int ntail     = (int)(n % 4);
    int* out            = (int*)d_out;             // int32 counts
    const int nbins     = out_size;

    // d_out is poisoned by the harness; zero it (graph-capture safe kernel).
    Model_zero_out<<<(out_size + TPB - 1) / TPB, TPB, 0, stream>>>(out, out_size);

    unsigned tiles  = (nvec + TPB - 1u) / TPB;
    unsigned blocks = tiles == 0u ? 1u : (tiles < 2048u ? tiles : 2048u);
    Model_moe_hist<<<blocks, TPB, 0, stream>>>(ids, nvec, ntail,
                                               expert_map, map_size, nbins, out);
}